// Cube_81501299409128
// MI455X (gfx1250) — compile-verified
//
#include <hip/hip_runtime.h>
#include <hip/hip_bf16.h>
#include <math.h>

typedef __attribute__((ext_vector_type(2))) float v2f;
typedef __attribute__((ext_vector_type(8))) float v8f;

#define RH 480
#define RW 640
#define R_NEAR 0.01f
#define R_FAR  10.0f
#define R_EPS  1e-9f

// workspace layout (floats)
#define WS_ROWSA 0            // 3 WMMAs * 16 rows * 4 coeffs = 192
#define WS_ZW    192          // 12 * 3 = 36
#define WS_VALID 228          // 12
#define WS_ATTR  240          // 12 * 3 * 9 = 324
// total 564 floats

// ---------------- setup: one thread computes all per-triangle constants ----------------
__global__ void cube_setup_kernel(const float* __restrict__ euler,
                                  const float* __restrict__ verts,
                                  const float* __restrict__ norms,
                                  const int*   __restrict__ tris,
                                  const float* __restrict__ colors,
                                  const float* __restrict__ eye,
                                  const float* __restrict__ center,
                                  const float* __restrict__ up,
                                  float* __restrict__ ws)
{
    if (threadIdx.x != 0 || blockIdx.x != 0) return;

    // rotation from euler angles
    float sx = sinf(euler[0]), sy = sinf(euler[1]), sz = sinf(euler[2]);
    float cx = cosf(euler[0]), cy = cosf(euler[1]), cz = cosf(euler[2]);
    float R[3][3] = {
        { cy*cz, sx*sy*cz - cx*sz, cx*sy*cz + sx*sz },
        { cy*sz, sx*sy*sz + cx*cz, cx*sy*sz - sx*cz },
        { -sy,   sx*cy,            cx*cy            }
    };

    float vw[8][3], nw[8][3];
    for (int v = 0; v < 8; ++v)
        for (int i = 0; i < 3; ++i) {
            vw[v][i] = verts[v*3+0]*R[i][0] + verts[v*3+1]*R[i][1] + verts[v*3+2]*R[i][2];
            nw[v][i] = norms[v*3+0]*R[i][0] + norms[v*3+1]*R[i][1] + norms[v*3+2]*R[i][2];
        }

    // look-at view matrix
    float f[3] = { center[0]-eye[0], center[1]-eye[1], center[2]-eye[2] };
    float fn = sqrtf(f[0]*f[0]+f[1]*f[1]+f[2]*f[2]);
    f[0]/=fn; f[1]/=fn; f[2]/=fn;
    float s3[3] = { f[1]*up[2]-f[2]*up[1], f[2]*up[0]-f[0]*up[2], f[0]*up[1]-f[1]*up[0] };
    float sn = sqrtf(s3[0]*s3[0]+s3[1]*s3[1]+s3[2]*s3[2]);
    s3[0]/=sn; s3[1]/=sn; s3[2]/=sn;
    float u3[3] = { s3[1]*f[2]-s3[2]*f[1], s3[2]*f[0]-s3[0]*f[2], s3[0]*f[1]-s3[1]*f[0] };
    float Rv[3][3] = { { s3[0], s3[1], s3[2] },
                       { u3[0], u3[1], u3[2] },
                       { -f[0], -f[1], -f[2] } };
    float view[4][4];
    for (int i = 0; i < 3; ++i) {
        view[i][0]=Rv[i][0]; view[i][1]=Rv[i][1]; view[i][2]=Rv[i][2];
        view[i][3] = -(Rv[i][0]*eye[0] + Rv[i][1]*eye[1] + Rv[i][2]*eye[2]);
    }
    view[3][0]=0.f; view[3][1]=0.f; view[3][2]=0.f; view[3][3]=1.f;

    // perspective
    const float aspect = (float)RW / (float)RH;
    const float focal  = 1.0f / tanf(0.3490658503988659f);   // radians(20)
    float proj[4][4] = {
        { focal/aspect, 0.f, 0.f, 0.f },
        { 0.f, focal, 0.f, 0.f },
        { 0.f, 0.f, (R_NEAR+R_FAR)/(R_NEAR-R_FAR), 2.0f*R_NEAR*R_FAR/(R_NEAR-R_FAR) },
        { 0.f, 0.f, -1.f, 0.f }
    };
    float PV[4][4];
    for (int i = 0; i < 4; ++i)
        for (int j = 0; j < 4; ++j) {
            float acc = 0.f;
            for (int k = 0; k < 4; ++k) acc += proj[i][k]*view[k][j];
            PV[i][j] = acc;
        }

    float clip[8][4];
    for (int v = 0; v < 8; ++v)
        for (int i = 0; i < 4; ++i)
            clip[v][i] = PV[i][0]*vw[v][0] + PV[i][1]*vw[v][1] + PV[i][2]*vw[v][2] + PV[i][3];

    // zero A rows (pad rows + k=3 must be 0)
    for (int i = 0; i < 192; ++i) ws[WS_ROWSA + i] = 0.f;

    for (int t = 0; t < 12; ++t) {
        int i0 = tris[3*t+0], i1 = tris[3*t+1], i2 = tris[3*t+2];
        float M[3][3] = {
            { clip[i0][0], clip[i1][0], clip[i2][0] },   // x row
            { clip[i0][1], clip[i1][1], clip[i2][1] },   // y row
            { clip[i0][3], clip[i1][3], clip[i2][3] }    // w row
        };
        float c00 =  (M[1][1]*M[2][2] - M[1][2]*M[2][1]);
        float c01 = -(M[1][0]*M[2][2] - M[1][2]*M[2][0]);
        float c02 =  (M[1][0]*M[2][1] - M[1][1]*M[2][0]);
        float det = M[0][0]*c00 + M[0][1]*c01 + M[0][2]*c02;
        bool valid = fabsf(det) > R_EPS;
        float inv[3][3];
        if (valid) {
            float id = 1.0f/det;
            inv[0][0] =  (M[1][1]*M[2][2]-M[1][2]*M[2][1])*id;
            inv[0][1] = -(M[0][1]*M[2][2]-M[0][2]*M[2][1])*id;
            inv[0][2] =  (M[0][1]*M[1][2]-M[0][2]*M[1][1])*id;
            inv[1][0] = -(M[1][0]*M[2][2]-M[1][2]*M[2][0])*id;
            inv[1][1] =  (M[0][0]*M[2][2]-M[0][2]*M[2][0])*id;
            inv[1][2] = -(M[0][0]*M[1][2]-M[0][2]*M[1][0])*id;
            inv[2][0] =  (M[1][0]*M[2][1]-M[1][1]*M[2][0])*id;
            inv[2][1] = -(M[0][0]*M[2][1]-M[0][1]*M[2][0])*id;
            inv[2][2] =  (M[0][0]*M[1][1]-M[0][1]*M[1][0])*id;
        } else {
            for (int i = 0; i < 3; ++i)
                for (int j = 0; j < 3; ++j)
                    inv[i][j] = (i==j) ? 1.f : 0.f;    // matches Msafe = I
        }
        ws[WS_VALID + t] = valid ? 1.f : 0.f;

        // A-matrix row packing: WMMA w = t/4, sub = t%4
        //   sub 0 -> rows 0..2, sub 1 -> rows 3..5, sub 2 -> rows 8..10, sub 3 -> rows 11..13
        int w = t >> 2;
        int sub = t & 3;
        int mbase = ((sub >> 1) * 8) + ((sub & 1) * 3);
        for (int comp = 0; comp < 3; ++comp) {
            float* row = ws + WS_ROWSA + (w*16 + mbase + comp)*4;
            row[0] = inv[comp][0];
            row[1] = inv[comp][1];
            row[2] = inv[comp][2];
            row[3] = 0.f;
        }

        int vid[3] = { i0, i1, i2 };
        for (int k = 0; k < 3; ++k) {
            ws[WS_ZW + t*3 + k] = clip[vid[k]][2] / clip[vid[k]][3];
            float* a = ws + WS_ATTR + (t*3 + k)*9;
            a[0] = vw[vid[k]][0]; a[1] = vw[vid[k]][1]; a[2] = vw[vid[k]][2];
            a[3] = nw[vid[k]][0]; a[4] = nw[vid[k]][1]; a[5] = nw[vid[k]][2];
            a[6] = colors[vid[k]*3+0]; a[7] = colors[vid[k]*3+1]; a[8] = colors[vid[k]*3+2];
        }
    }
}

// ---------------- per-pixel raster: each wave rasterizes 16 pixels via 3 WMMAs ----------------
__global__ __launch_bounds__(256) void cube_raster_kernel(const float* __restrict__ ws,
                                                          const float* __restrict__ lightpos,
                                                          const float* __restrict__ lightint,
                                                          float* __restrict__ out)
{
    const int lane  = threadIdx.x & 31;
    const int wave  = (blockIdx.x * blockDim.x + threadIdx.x) >> 5;   // 0..19199
    const int y     = wave / (RW/16);
    const int xbase = (wave % (RW/16)) * 16;
    const int half  = lane >> 4;
    const int pcol  = lane & 15;
    const int x     = xbase + pcol;

    // B operand: K=0,1 (px,py) in lanes 0-15; K=2,3 (1,0) in lanes 16-31
    v2f bop;
    {
        float px = ((float)x + 0.5f) * (2.0f/(float)RW) - 1.0f;
        float py = 1.0f - ((float)y + 0.5f) * (2.0f/(float)RH);
        bop.x = half ? 1.0f : px;
        bop.y = half ? 0.0f : py;
    }

    const float* rowsA  = ws + WS_ROWSA;
    const float* zw     = ws + WS_ZW;
    const float* validf = ws + WS_VALID;
    const int kbase = half ? 2 : 0;

    float bestD = 3.0e38f;
    int   bestId = 0x7fffffff;
    float bn0 = 0.f, bn1 = 0.f, bn2 = 0.f;
    bool  anyIn = false;

    auto proc = [&](int t, float b0, float b1, float b2) {
        bool inside = (validf[t] > 0.5f) && (b0 >= 0.f) && (b1 >= 0.f) && (b2 >= 0.f);
        float bsum = b0 + b1 + b2;
        float s = (fabsf(bsum) > R_EPS) ? bsum : 1.0f;
        float is = 1.0f / s;
        float n0 = b0*is, n1 = b1*is, n2 = b2*is;
        float depth = inside ? (n0*zw[t*3+0] + n1*zw[t*3+1] + n2*zw[t*3+2]) : 1.0e9f;
        anyIn = anyIn || inside;
        if (depth < bestD) { bestD = depth; bestId = t; bn0 = n0; bn1 = n1; bn2 = n2; }
    };

    #pragma unroll
    for (int w = 0; w < 3; ++w) {
        // A operand: lane supplies row (lane%16); K={0,1} for lanes 0-15, K={2,3} for lanes 16-31
        const float* arow = rowsA + (w*16 + pcol)*4;
        v2f aop; aop.x = arow[kbase]; aop.y = arow[kbase+1];
        v8f c = {};
        v8f d = __builtin_amdgcn_wmma_f32_16x16x4_f32(
            /*neg_a=*/false, aop, /*neg_b=*/false, bop,
            /*c_mod=*/(short)0, c, /*reuse_a=*/false, /*reuse_b=*/false);
        // lane holds D rows {0..7} (lanes 0-15) or {8..15} (lanes 16-31) of column pcol
        int t0 = 4*w + 2*half;          // triangles processed in increasing global id per lane
        proc(t0 + 0, d[0], d[1], d[2]);
        proc(t0 + 1, d[3], d[4], d[5]);
    }

    // combine lane pair (L, L+16): min-depth, tie -> lower triangle id (jnp.argmin semantics)
    float oD  = __shfl_xor(bestD, 16);
    int   oId = __shfl_xor(bestId, 16);
    float o0  = __shfl_xor(bn0, 16);
    float o1  = __shfl_xor(bn1, 16);
    float o2  = __shfl_xor(bn2, 16);
    int   oIn = __shfl_xor((int)anyIn, 16);
    if (oD < bestD || (oD == bestD && oId < bestId)) {
        bestD = oD; bestId = oId; bn0 = o0; bn1 = o1; bn2 = o2;
    }
    bool hit = anyIn || (oIn != 0);

    // interpolate 9 attributes of winning triangle
    float A[9];
    const float* ta = ws + WS_ATTR + bestId*27;
    #pragma unroll
    for (int c9 = 0; c9 < 9; ++c9)
        A[c9] = bn0*ta[c9] + bn1*ta[9+c9] + bn2*ta[18+c9];

    const int pix = y*RW + x;
    float* outR = out;                         // rendered: H*W*4
    float* outG = out + (size_t)RH*RW*4;       // geometry: H*W*9

    if (half == 0) {
        float4 r = make_float4(0.f, 0.f, 0.f, 0.f);
        if (hit) {
            float nx = A[3], ny = A[4], nz = A[5];
            float nn = sqrtf(nx*nx + ny*ny + nz*nz) + R_EPS;
            nx /= nn; ny /= nn; nz /= nn;
            float lx = lightpos[0]-A[0], ly = lightpos[1]-A[1], lz = lightpos[2]-A[2];
            float ln = sqrtf(lx*lx + ly*ly + lz*lz) + R_EPS;
            lx /= ln; ly /= ln; lz /= ln;
            float lam = fmaxf(0.f, nx*lx + ny*ly + nz*lz);
            r.x = lam * lightint[0] * A[6];
            r.y = lam * lightint[1] * A[7];
            r.z = lam * lightint[2] * A[8];
            r.w = 1.0f;
        }
        *reinterpret_cast<float4*>(outR + (size_t)pix*4) = r;
    } else {
        float* gp = outG + (size_t)pix*9;
        #pragma unroll
        for (int c9 = 0; c9 < 9; ++c9)
            gp[c9] = hit ? A[c9] : 0.0f;
    }
}

extern "C" void kernel_launch(void* const* d_in, const int* in_sizes, int n_in,
                              void* d_out, int out_size, void* d_ws, size_t ws_size,
                              hipStream_t stream) {
    const float* euler   = (const float*)d_in[0];
    const float* verts   = (const float*)d_in[1];
    const float* norms   = (const float*)d_in[2];
    const int*   tris    = (const int*)  d_in[3];
    const float* colors  = (const float*)d_in[4];
    const float* eye     = (const float*)d_in[5];
    const float* center  = (const float*)d_in[6];
    const float* up      = (const float*)d_in[7];
    const float* lpos    = (const float*)d_in[8];
    const float* lint    = (const float*)d_in[9];
    float* out = (float*)d_out;
    float* ws  = (float*)d_ws;

    cube_setup_kernel<<<1, 32, 0, stream>>>(euler, verts, norms, tris, colors,
                                            eye, center, up, ws);

    // 480*640 pixels / 16 per wave = 19200 waves; 8 waves per 256-thread block -> 2400 blocks
    cube_raster_kernel<<<2400, 256, 0, stream>>>(ws, lpos, lint, out);
}